// GraphEncoder_69681549410865
// MI455X (gfx1250) — compile-verified
//
#include <hip/hip_runtime.h>
#include <hip/hip_bf16.h>

#define NN  100000   // nodes
#define EE  1600000  // edges
#define DIN 128
#define H   64
#define LL  4
#define GG  64

typedef float v2f __attribute__((ext_vector_type(2)));
typedef float v8f __attribute__((ext_vector_type(8)));

// ---------------------------------------------------------------------------
// Kernel 1: node encoder  h = x @ node_w^T + node_b   ([N,128] x [128,64])
// One wave computes a 16(M) x 64(N) strip with V_WMMA_F32_16X16X4_F32.
// ---------------------------------------------------------------------------
__global__ __launch_bounds__(256) void k_encoder(const float* __restrict__ x,
                                                 const float* __restrict__ w,
                                                 const float* __restrict__ bias,
                                                 float* __restrict__ h)
{
    const int lane = threadIdx.x & 31;
    const int wv   = threadIdx.x >> 5;
    const int row0 = (blockIdx.x * 8 + wv) * 16;
    if (row0 >= NN) return;                       // wave-uniform

    const int half = lane >> 4;                   // 0: K lo pair, 1: K hi pair
    const int ln   = lane & 15;

    v8f acc[4] = {};                              // 4 N-tiles of 16x16 f32

    int rA = row0 + ln;                           // A rows: M = ln (both halves)
    if (rA >= NN) rA = NN - 1;                    // clamp (stores guarded later)
    const float* arow = x + (size_t)rA * DIN;

    for (int k = 0; k < DIN; k += 4) {
        const int ka = k + half * 2;
        v2f a; a.x = arow[ka]; a.y = arow[ka + 1];
#pragma unroll
        for (int t = 0; t < 4; ++t) {
            const int n = t * 16 + ln;            // B col, B[k][n] = w[n*DIN+k]
            v2f bb; bb.x = w[(size_t)n * DIN + ka];
                    bb.y = w[(size_t)n * DIN + ka + 1];
            acc[t] = __builtin_amdgcn_wmma_f32_16x16x4_f32(
                false, a, false, bb, (short)0, acc[t], false, false);
        }
    }
#pragma unroll
    for (int t = 0; t < 4; ++t) {
        const int col = t * 16 + ln;
        const float bv = bias[col];
#pragma unroll
        for (int q = 0; q < 8; ++q) {
            const int row = row0 + q + half * 8;
            if (row < NN) h[(size_t)row * H + col] = acc[t][q] + bv;
        }
    }
}

// ---------------------------------------------------------------------------
// Kernel 2: hn = relu(layernorm(h)) ; agg = 0   (one wave per node row)
// ---------------------------------------------------------------------------
__global__ __launch_bounds__(256) void k_prenorm(const float* __restrict__ h,
                                                 const float* __restrict__ g,
                                                 const float* __restrict__ b,
                                                 float* __restrict__ hn,
                                                 float* __restrict__ agg)
{
    const int row  = blockIdx.x * (blockDim.x >> 5) + (threadIdx.x >> 5);
    const int lane = threadIdx.x & 31;
    if (row >= NN) return;

    const float2 v = ((const float2*)(h + (size_t)row * H))[lane];
    float s = v.x + v.y;
#pragma unroll
    for (int m = 16; m >= 1; m >>= 1) s += __shfl_xor(s, m, 32);
    const float mu = s * (1.0f / 64.0f);
    const float dx = v.x - mu, dy = v.y - mu;
    float vs = dx * dx + dy * dy;
#pragma unroll
    for (int m = 16; m >= 1; m >>= 1) vs += __shfl_xor(vs, m, 32);
    const float inv = rsqrtf(vs * (1.0f / 64.0f) + 1e-5f);

    const int c0 = lane * 2;
    float2 o;
    o.x = fmaxf(dx * inv * g[c0]     + b[c0],     0.0f);
    o.y = fmaxf(dy * inv * g[c0 + 1] + b[c0 + 1], 0.0f);
    ((float2*)(hn + (size_t)row * H))[lane] = o;
    ((float2*)(agg + (size_t)row * H))[lane] = make_float2(0.0f, 0.0f);
}

// ---------------------------------------------------------------------------
// Kernel 3: edge scatter-add  agg[dst] += hn[src]   (lane = (edge, 4 cols))
// ---------------------------------------------------------------------------
__global__ __launch_bounds__(256) void k_scatter(const int* __restrict__ src,
                                                 const int* __restrict__ dst,
                                                 const float* __restrict__ hn,
                                                 float* __restrict__ agg)
{
    const long long gidx = (long long)blockIdx.x * blockDim.x + threadIdx.x;
    const int e  = (int)(gidx >> 4);
    const int c4 = ((int)gidx & 15) * 4;
    if (e >= EE) return;
    const int s = src[e], d = dst[e];
    const float4 v = *(const float4*)(hn + (size_t)s * H + c4);
    float* p = agg + (size_t)d * H + c4;
    atomicAdd(p + 0, v.x);
    atomicAdd(p + 1, v.y);
    atomicAdd(p + 2, v.z);
    atomicAdd(p + 3, v.w);
}

// ---------------------------------------------------------------------------
// Kernel 4: fused GIN MLP:
//   z  = (1+eps)*hn + agg
//   t  = relu(bn1(z @ w1^T + b1))     (stored to per-wave LDS tile)
//   t2 = relu(bn2(t @ w2^T + b2))
//   h += t2
// One wave per 16-row strip; both GEMMs via WMMA f32 16x16x4.
// ---------------------------------------------------------------------------
__global__ __launch_bounds__(256) void k_gin_mlp(
    const float* __restrict__ hn, const float* __restrict__ agg,
    const float* __restrict__ eps, int layer,
    const float* __restrict__ w1, const float* __restrict__ b1,
    const float* __restrict__ g1, const float* __restrict__ bb1,
    const float* __restrict__ m1, const float* __restrict__ v1,
    const float* __restrict__ w2, const float* __restrict__ b2,
    const float* __restrict__ g2, const float* __restrict__ bb2,
    const float* __restrict__ m2, const float* __restrict__ v2,
    float* __restrict__ h)
{
    const int lane = threadIdx.x & 31;
    const int wv   = threadIdx.x >> 5;
    const int row0 = (blockIdx.x * 8 + wv) * 16;
    if (row0 >= NN) return;

    const int half = lane >> 4;
    const int ln   = lane & 15;
    const float ep = 1.0f + eps[layer];

    __shared__ float tb[8][16][H + 4];            // padded: conflict-free reads
    float (*tt)[H + 4] = tb[wv];

    // ----- GEMM 1: A built on the fly ------------------------------------
    v8f acc[4] = {};
    int rA = row0 + ln; if (rA >= NN) rA = NN - 1;
    const float* hnr = hn  + (size_t)rA * H;
    const float* agr = agg + (size_t)rA * H;

    for (int k = 0; k < H; k += 4) {
        const int ka = k + half * 2;
        v2f a;
        a.x = fmaf(ep, hnr[ka],     agr[ka]);
        a.y = fmaf(ep, hnr[ka + 1], agr[ka + 1]);
#pragma unroll
        for (int t = 0; t < 4; ++t) {
            const int n = t * 16 + ln;
            v2f bb; bb.x = w1[(size_t)n * H + ka];
                    bb.y = w1[(size_t)n * H + ka + 1];
            acc[t] = __builtin_amdgcn_wmma_f32_16x16x4_f32(
                false, a, false, bb, (short)0, acc[t], false, false);
        }
    }

    // ----- bn1 + relu -> LDS tile (16 x 64) ------------------------------
#pragma unroll
    for (int t = 0; t < 4; ++t) {
        const int c  = t * 16 + ln;
        const float sc = g1[c] * rsqrtf(v1[c] + 1e-5f);
        const float sh = bb1[c] - m1[c] * sc;
#pragma unroll
        for (int q = 0; q < 8; ++q) {
            const int rr = q + half * 8;
            tt[rr][c] = fmaxf(fmaf(acc[t][q] + b1[c], sc, sh), 0.0f);
        }
    }

    // ----- GEMM 2: A from LDS --------------------------------------------
    v8f acc2[4] = {};
    for (int k = 0; k < H; k += 4) {
        const int ka = k + half * 2;
        v2f a; a.x = tt[ln][ka]; a.y = tt[ln][ka + 1];
#pragma unroll
        for (int t = 0; t < 4; ++t) {
            const int n = t * 16 + ln;
            v2f bb; bb.x = w2[(size_t)n * H + ka];
                    bb.y = w2[(size_t)n * H + ka + 1];
            acc2[t] = __builtin_amdgcn_wmma_f32_16x16x4_f32(
                false, a, false, bb, (short)0, acc2[t], false, false);
        }
    }

    // ----- bn2 + relu + residual add -------------------------------------
#pragma unroll
    for (int t = 0; t < 4; ++t) {
        const int c  = t * 16 + ln;
        const float sc = g2[c] * rsqrtf(v2[c] + 1e-5f);
        const float sh = bb2[c] - m2[c] * sc;
#pragma unroll
        for (int q = 0; q < 8; ++q) {
            const int row = row0 + q + half * 8;
            if (row < NN) {
                const float val = fmaxf(fmaf(acc2[t][q] + b2[c], sc, sh), 0.0f);
                h[(size_t)row * H + c] += val;
            }
        }
    }
}

// ---------------------------------------------------------------------------
// Pooling kernels
// ---------------------------------------------------------------------------
__global__ __launch_bounds__(256) void k_zero_pool(float* __restrict__ pool,
                                                   float* __restrict__ cnt)
{
    const int i = blockIdx.x * blockDim.x + threadIdx.x;
    if (i < GG * H) pool[i] = 0.0f;
    if (i < GG)     cnt[i]  = 0.0f;
}

__global__ __launch_bounds__(256) void k_pool(const float* __restrict__ h,
                                              const int* __restrict__ batch,
                                              float* __restrict__ pool,
                                              float* __restrict__ cnt)
{
    const long long gidx = (long long)blockIdx.x * blockDim.x + threadIdx.x;
    const int n  = (int)(gidx >> 4);
    const int c4 = ((int)gidx & 15) * 4;
    if (n >= NN) return;
    const int b = batch[n];
    const float4 v = *(const float4*)(h + (size_t)n * H + c4);
    float* p = pool + (size_t)b * H + c4;
    atomicAdd(p + 0, v.x);
    atomicAdd(p + 1, v.y);
    atomicAdd(p + 2, v.z);
    atomicAdd(p + 3, v.w);
    if (c4 == 0) atomicAdd(cnt + b, 1.0f);
}

__global__ __launch_bounds__(256) void k_final(const float* __restrict__ pool,
                                               const float* __restrict__ cnt,
                                               float* __restrict__ out)
{
    const int i = blockIdx.x * blockDim.x + threadIdx.x;
    if (i >= GG * H) return;
    const int g = i / H;
    out[i] = pool[i] / fmaxf(cnt[g], 1.0f);
}

// ---------------------------------------------------------------------------
extern "C" void kernel_launch(void* const* d_in, const int* in_sizes, int n_in,
                              void* d_out, int out_size, void* d_ws, size_t ws_size,
                              hipStream_t stream)
{
    const float* x      = (const float*)d_in[0];
    const float* node_w = (const float*)d_in[1];
    const float* node_b = (const float*)d_in[2];
    const float* ln_g   = (const float*)d_in[3];
    const float* ln_b   = (const float*)d_in[4];
    const float* eps    = (const float*)d_in[5];
    const float* w1     = (const float*)d_in[6];
    const float* b1     = (const float*)d_in[7];
    const float* bn1_g  = (const float*)d_in[8];
    const float* bn1_b  = (const float*)d_in[9];
    const float* bn1_m  = (const float*)d_in[10];
    const float* bn1_v  = (const float*)d_in[11];
    const float* w2     = (const float*)d_in[12];
    const float* b2     = (const float*)d_in[13];
    const float* bn2_g  = (const float*)d_in[14];
    const float* bn2_b  = (const float*)d_in[15];
    const float* bn2_m  = (const float*)d_in[16];
    const float* bn2_v  = (const float*)d_in[17];
    const int*   ei     = (const int*)d_in[18];
    const int*   batch  = (const int*)d_in[19];
    float* out = (float*)d_out;

    float* ws   = (float*)d_ws;
    float* hn   = ws;                              // [N,H]
    float* agg  = ws + (size_t)1 * NN * H;         // [N,H]
    float* h    = ws + (size_t)2 * NN * H;         // [N,H]
    float* pool = ws + (size_t)3 * NN * H;         // [G,H]
    float* cnt  = pool + GG * H;                   // [G]

    const dim3 blk(256);
    const int grid_gemm    = (NN + 127) / 128;     // 8 waves x 16 rows
    const int grid_prenorm = (NN + 7) / 8;         // 8 waves x 1 row
    const int grid_scatter = (EE * 16 + 255) / 256;
    const int grid_pool    = (NN * 16 + 255) / 256;

    k_encoder<<<grid_gemm, blk, 0, stream>>>(x, node_w, node_b, h);

    for (int i = 0; i < LL; ++i) {
        k_prenorm<<<grid_prenorm, blk, 0, stream>>>(h, ln_g + i * H, ln_b + i * H, hn, agg);
        k_scatter<<<grid_scatter, blk, 0, stream>>>(ei, ei + EE, hn, agg);
        k_gin_mlp<<<grid_gemm, blk, 0, stream>>>(
            hn, agg, eps, i,
            w1 + (size_t)i * H * H, b1 + i * H,
            bn1_g + i * H, bn1_b + i * H, bn1_m + i * H, bn1_v + i * H,
            w2 + (size_t)i * H * H, b2 + i * H,
            bn2_g + i * H, bn2_b + i * H, bn2_m + i * H, bn2_v + i * H,
            h);
    }

    k_zero_pool<<<(GG * H + 255) / 256, blk, 0, stream>>>(pool, cnt);
    k_pool<<<grid_pool, blk, 0, stream>>>(h, batch, pool, cnt);
    k_final<<<(GG * H + 255) / 256, blk, 0, stream>>>(pool, cnt, out);
}